// InvariantMessagePasser_21474836480309
// MI455X (gfx1250) — compile-verified
//
#include <hip/hip_runtime.h>

typedef __attribute__((ext_vector_type(2))) float v2f;
typedef __attribute__((ext_vector_type(8))) float v8f;

#define N_ATOMS 10000
#define N_EDGES 250000
#define CHUNK   256

// output layout (flat f32, concatenated in return order)
#define OUT_OFF1 (N_ATOMS * 256)
#define OUT_OFF2 (OUT_OFF1 + N_ATOMS * 3 * 192)
#define OUT_OFF3 (OUT_OFF2 + N_ATOMS * 5 * 128)

// ---------------- phase 1: counting sort of edges by center ----------------

__global__ void zero_i32(int* __restrict__ p, int n) {
  int i = blockIdx.x * blockDim.x + threadIdx.x;
  if (i < n) p[i] = 0;
}

__global__ void hist_centers(const int* __restrict__ centers,
                             int* __restrict__ counts, int n) {
  int i = blockIdx.x * blockDim.x + threadIdx.x;
  if (i < n) atomicAdd(&counts[centers[i]], 1);
}

__global__ __launch_bounds__(1024)
void scan_counts(const int* __restrict__ counts, int* __restrict__ offsets, int n) {
  __shared__ int buf[1024];
  __shared__ int running;
  int tid = threadIdx.x;
  if (tid == 0) running = 0;
  __syncthreads();
  for (int base = 0; base < n; base += 1024) {
    int i = base + tid;
    int v = (i < n) ? counts[i] : 0;
    buf[tid] = v;
    __syncthreads();
    for (int off = 1; off < 1024; off <<= 1) {
      int t = (tid >= off) ? buf[tid - off] : 0;
      __syncthreads();
      buf[tid] += t;
      __syncthreads();
    }
    int incl = buf[tid];
    if (i < n) offsets[i] = running + incl - v;  // exclusive prefix
    int total = buf[1023];
    __syncthreads();
    if (tid == 0) running += total;
    __syncthreads();
  }
}

__global__ void scatter_perm(const int* __restrict__ centers,
                             const int* __restrict__ offsets,
                             int* __restrict__ cursor,
                             int* __restrict__ perm, int n) {
  int i = blockIdx.x * blockDim.x + threadIdx.x;
  if (i < n) {
    int c = centers[i];
    int pos = offsets[c] + atomicAdd(&cursor[c], 1);
    perm[pos] = i;
  }
}

// ---------------- phase 2: per-atom density via f32 WMMA outer products ----
//
// One block (8 wave32) per atom. Packed M rows: row = l*l + m  (16 rows total).
// 40 (l, ktile) output tiles; wave w owns tiles w*5 .. w*5+4.
// dens_l[atom] = sum over edge groups of A(16x4 sh) x B(4x16 rb*emb)
// via V_WMMA_F32_16X16X4_F32, accumulating in f32.
// Tile metadata is forced scalar (readfirstlane) and hoisted out of the
// hot loop so the inner body is loads + mul + wmma with full EXEC.

__global__ __launch_bounds__(256)
void impass_density(const float* __restrict__ sh0, const float* __restrict__ sh1,
                    const float* __restrict__ sh2, const float* __restrict__ sh3,
                    const float* __restrict__ rb0, const float* __restrict__ rb1,
                    const float* __restrict__ rb2, const float* __restrict__ rb3,
                    const float* __restrict__ emb, const int* __restrict__ neighbors,
                    const int* __restrict__ offsets, const int* __restrict__ counts,
                    const int* __restrict__ perm, float* __restrict__ out) {
  __shared__ int   s_raw[CHUNK];
  __shared__ int   s_eid[CHUNK];
  __shared__ int   s_nbr[CHUNK];
  __shared__ float s_sh[CHUNK * 16];

  const int atom = blockIdx.x;
  const int tid  = threadIdx.x;
  const int lane = tid & 31;
  const int nlo  = lane & 15;
  const bool hiHalf = (lane >= 16);
  // wave id as a true scalar (uniform within wave) -> scalar tile metadata
  const int wave = __builtin_amdgcn_readfirstlane((int)threadIdx.x) >> 5;

  const int start = offsets[atom];
  const int len   = counts[atom];

  v8f acc[5] = {};  // zero-init accumulators

  // per-tile metadata, hoisted: scalar l/kt/kl/rb pointer, vector column kv
  int ls[5], kts[5], kls[5], kv[5];
  const float* rbp[5];
#pragma unroll
  for (int s = 0; s < 5; ++s) {
    int t = wave * 5 + s;  // scalar
    int l, kt;
    if (t < 16)      { l = 0; kt = t; }
    else if (t < 28) { l = 1; kt = t - 16; }
    else if (t < 36) { l = 2; kt = t - 28; }
    else             { l = 3; kt = t - 36; }
    ls[s] = l; kts[s] = kt;
    if (l == 0)      { rbp[s] = rb0; kls[s] = 256; }
    else if (l == 1) { rbp[s] = rb1; kls[s] = 192; }
    else if (l == 2) { rbp[s] = rb2; kls[s] = 128; }
    else             { rbp[s] = rb3; kls[s] = 64;  }
    kv[s] = kt * 16 + nlo;  // per-lane column index (same k for rb and emb)
  }

  for (int c0 = 0; c0 < len; c0 += CHUNK) {
    int clen = len - c0; if (clen > CHUNK) clen = CHUNK;

    // load this chunk's edge ids
    if (tid < clen) s_raw[tid] = perm[start + c0 + tid];
    __syncthreads();
    // deterministic rank sort (edge ids unique)
    if (tid < clen) {
      int my = s_raw[tid];
      int rank = 0;
      for (int j = 0; j < clen; ++j) rank += (s_raw[j] < my) ? 1 : 0;
      s_eid[rank] = my;
    }
    __syncthreads();
    // stage neighbor ids + packed 16-row sh per edge
    if (tid < clen) {
      int e = s_eid[tid];
      s_nbr[tid] = neighbors[e];
      float* row = &s_sh[tid * 16];
      row[0] = sh0[e];
#pragma unroll
      for (int m = 0; m < 3; ++m) row[1 + m] = sh1[e * 3 + m];
#pragma unroll
      for (int m = 0; m < 5; ++m) row[4 + m] = sh2[e * 5 + m];
#pragma unroll
      for (int m = 0; m < 7; ++m) row[9 + m] = sh3[e * 7 + m];
    }
    __syncthreads();

    int ngroups = (clen + 3) >> 2;
    for (int g = 0; g < ngroups; ++g) {
      int base = g * 4;
      int i0 = base + (hiHalf ? 2 : 0);  // reg0 edge: 0 (lo lanes) / 2 (hi lanes)
      int i1 = i0 + 1;                   // reg1 edge: 1 / 3
      float m0 = (i0 < clen) ? 1.0f : 0.0f;
      float m1 = (i1 < clen) ? 1.0f : 0.0f;
      int p0 = i0 < clen - 1 ? i0 : clen - 1;
      int p1 = i1 < clen - 1 ? i1 : clen - 1;
      int e0 = s_eid[p0], e1 = s_eid[p1];
      int eb0 = s_nbr[p0] * 256;  // emb row base
      int eb1 = s_nbr[p1] * 256;

      // A operand: sh rows (shared across all tiles of this wave)
      v2f a;
      a.x = s_sh[p0 * 16 + nlo];
      a.y = s_sh[p1 * 16 + nlo];

#pragma unroll
      for (int s = 0; s < 5; ++s) {
        const float* rb = rbp[s];
        const int kl = kls[s];
        const int k  = kv[s];
        v2f b;
        b.x = rb[e0 * kl + k] * emb[eb0 + k] * m0;
        b.y = rb[e1 * kl + k] * emb[eb1 + k] * m1;
        acc[s] = __builtin_amdgcn_wmma_f32_16x16x4_f32(
            /*neg_a=*/false, a, /*neg_b=*/false, b,
            /*c_mod=*/(short)0, acc[s], /*reuse_a=*/false, /*reuse_b=*/false);
      }
    }
    __syncthreads();
  }

  // ------- store: extract rows (row = l*l + m) from D layout, plain stores ----
#pragma unroll
  for (int s = 0; s < 5; ++s) {
    int l = ls[s];               // scalar -> s_cbranch, no exec churn
    int k = kts[s] * 16 + nlo;
    if (l == 0) {
      if (!hiHalf) out[atom * 256 + k] = acc[s][0];           // row 0 -> v0, lo
    } else if (l == 1) {
      if (!hiHalf) {                                           // rows 1..3 -> v1..v3, lo
        float* o = out + OUT_OFF1 + atom * 3 * 192 + k;
        o[0 * 192] = acc[s][1]; o[1 * 192] = acc[s][2]; o[2 * 192] = acc[s][3];
      }
    } else if (l == 2) {
      float* o = out + OUT_OFF2 + atom * 5 * 128 + k;
      if (!hiHalf) {                                           // rows 4..7 -> v4..v7, lo
        o[0 * 128] = acc[s][4]; o[1 * 128] = acc[s][5];
        o[2 * 128] = acc[s][6]; o[3 * 128] = acc[s][7];
      } else {                                                 // row 8 -> v0, hi
        o[4 * 128] = acc[s][0];
      }
    } else {
      if (hiHalf) {                                            // rows 9..15 -> v1..v7, hi
        float* o = out + OUT_OFF3 + atom * 7 * 64 + k;
#pragma unroll
        for (int m = 0; m < 7; ++m) o[m * 64] = acc[s][m + 1];
      }
    }
  }
}

// ---------------- launcher ----------------

extern "C" void kernel_launch(void* const* d_in, const int* in_sizes, int n_in,
                              void* d_out, int out_size, void* d_ws, size_t ws_size,
                              hipStream_t stream) {
  // setup_inputs order: sh_0, rb_0, sh_1, rb_1, sh_2, rb_2, sh_3, rb_3, emb, centers, neighbors
  const float* sh0 = (const float*)d_in[0];
  const float* rb0 = (const float*)d_in[1];
  const float* sh1 = (const float*)d_in[2];
  const float* rb1 = (const float*)d_in[3];
  const float* sh2 = (const float*)d_in[4];
  const float* rb2 = (const float*)d_in[5];
  const float* sh3 = (const float*)d_in[6];
  const float* rb3 = (const float*)d_in[7];
  const float* emb = (const float*)d_in[8];
  const int* centers   = (const int*)d_in[9];
  const int* neighbors = (const int*)d_in[10];
  float* out = (float*)d_out;

  // workspace carve-out (~1.12 MB)
  int* counts  = (int*)d_ws;
  int* cursor  = counts + N_ATOMS;
  int* offsets = cursor + N_ATOMS;
  int* perm    = offsets + N_ATOMS;

  zero_i32<<<(2 * N_ATOMS + 255) / 256, 256, 0, stream>>>(counts, 2 * N_ATOMS);
  hist_centers<<<(N_EDGES + 255) / 256, 256, 0, stream>>>(centers, counts, N_EDGES);
  scan_counts<<<1, 1024, 0, stream>>>(counts, offsets, N_ATOMS);
  scatter_perm<<<(N_EDGES + 255) / 256, 256, 0, stream>>>(centers, offsets, cursor, perm, N_EDGES);
  impass_density<<<N_ATOMS, 256, 0, stream>>>(sh0, sh1, sh2, sh3, rb0, rb1, rb2, rb3,
                                              emb, neighbors, offsets, counts, perm, out);
}